// AnchorPostProcess_15719580304314
// MI455X (gfx1250) — compile-verified
//
#include <hip/hip_runtime.h>

// ---------------- problem constants (mirror reference) ----------------
#define Bc 8
#define Ac 6
#define Cc 16
#define Hc 160
#define Wc 160
#define HWc (Hc * Wc)          // 25600
#define Nc (Ac * HWc)          // 153600 boxes per image
#define NCH (Ac * (4 + Cc))    // 120 channels
#define PREK 2000
#define KPAD 2048
#define POSTK 100
#define SCORE_THR 0.05f
#define IOU_THR 0.5f
#define IMG_WF 1280.0f
#define IMG_HF 1280.0f
#define DW_CLAMP 4.0f

typedef float v2f __attribute__((ext_vector_type(2)));
typedef float v8f __attribute__((ext_vector_type(8)));

// ---------------- workspace layout (bytes) ----------------
// zero-init region (single hipMemsetAsync):
#define OFF_HIST   ((size_t)0)                        // u32 [8][1024]   = 32768
#define OFF_FINE   ((size_t)32768)                    // u32 [8][4096]   = 131072
#define OFF_CNT    ((size_t)(OFF_FINE + 131072))      // u32 [8][2]      = 64
#define OFF_THR    ((size_t)(OFF_CNT + 64))           // u32 [8][4]      = 128
#define OFF_CAND   ((size_t)164096)                   // u64 [8][2048]   = 131072 (8B aligned)
#define ZERO_BYTES ((size_t)(OFF_CAND + 131072))      // 295168
// non-zeroed scratch:
#define OFF_BOXES  ((size_t)295168)                   // f32 [8][N][4]   = 19660800 (16B aligned)
#define OFF_KEY    ((size_t)(OFF_BOXES + 19660800))   // u32 [8][N]      = 4915200
#define OFF_CLS    ((size_t)(OFF_KEY + 4915200))      // i32 [8][N]      = 4915200
#define OFF_SSC    ((size_t)(OFF_CLS + 4915200))      // f32 [8][2048]   = 65536
#define OFF_SBOX   ((size_t)(OFF_SSC + 65536))        // f32 [8][2048][4]= 262144 (16B aligned)
#define OFF_SCLS   ((size_t)(OFF_SBOX + 262144))      // i32 [8][2048]   = 65536
#define OFF_MASK   ((size_t)(OFF_SCLS + 65536))       // u32 [8][2048][64]= 4194304
// total ~34.4 MB

// =====================================================================
// K1: decode boxes + score/cls, build per-image coarse histogram.
// grid = 8*600 blocks of 256 threads; each thread = one (b,a,h,w) box.
// =====================================================================
__global__ void __launch_bounds__(256) decode_kernel(
    const float* __restrict__ anc, const float* __restrict__ ho,
    float* __restrict__ boxes, unsigned* __restrict__ key,
    int* __restrict__ clsArr, unsigned* __restrict__ hist) {
  __shared__ unsigned lh[1024];
  for (int i = threadIdx.x; i < 1024; i += 256) lh[i] = 0u;
  __syncthreads();

  const int blocksPerImg = Nc / 256;  // 600
  const int b = blockIdx.x / blocksPerImg;
  const int n = (blockIdx.x % blocksPerImg) * 256 + threadIdx.x;
  const int a = n / HWc;
  const int hw = n - a * HWc;

  const size_t base = ((size_t)(b * NCH + a * (4 + Cc))) * HWc + hw;
  const float dx = ho[base];
  const float dy = ho[base + (size_t)HWc];
  const float dw = fminf(ho[base + 2 * (size_t)HWc], DW_CLAMP);
  const float dh = fminf(ho[base + 3 * (size_t)HWc], DW_CLAMP);

  float ml = -3.0e38f;
  int mc = 0;
#pragma unroll
  for (int c = 0; c < Cc; ++c) {
    float l = ho[base + (size_t)(4 + c) * HWc];
    if (l > ml) { ml = l; mc = c; }
  }
  const float score = 1.0f / (1.0f + expf(-ml));  // sigmoid monotone => max/argmax preserved

  const size_t ab = (size_t)(a * 4) * HWc + hw;
  const float ax1 = anc[ab], ay1 = anc[ab + HWc];
  const float ax2 = anc[ab + 2 * (size_t)HWc], ay2 = anc[ab + 3 * (size_t)HWc];
  const float aw = ax2 - ax1, ah = ay2 - ay1;
  const float acx = ax1 + 0.5f * aw, acy = ay1 + 0.5f * ah;
  const float gcx = acx + dx * aw, gcy = acy + dy * ah;
  const float gw = aw * expf(dw), gh = ah * expf(dh);
  const float x1 = fminf(fmaxf(gcx - 0.5f * gw, 0.0f), IMG_WF);
  const float y1 = fminf(fmaxf(gcy - 0.5f * gh, 0.0f), IMG_HF);
  const float x2 = fminf(fmaxf(gcx + 0.5f * gw, 0.0f), IMG_WF);
  const float y2 = fminf(fmaxf(gcy + 0.5f * gh, 0.0f), IMG_HF);

  const bool valid = (score >= SCORE_THR) && ((x2 - x1) >= 2.0f) && ((y2 - y1) >= 2.0f);
  const unsigned k = valid ? __float_as_uint(score) : 0u;  // score in (0,1): bits monotone

  const size_t o = (size_t)b * Nc + n;
  ((float4*)boxes)[o] = make_float4(x1, y1, x2, y2);
  key[o] = k;
  clsArr[o] = mc;
  if (valid) atomicAdd(&lh[k >> 20], 1u);
  __syncthreads();
  for (int i = threadIdx.x; i < 1024; i += 256)
    if (lh[i]) atomicAdd(&hist[b * 1024 + i], lh[i]);
}

// =====================================================================
// K2a: radix-select pass 1 — coarse threshold bin per image.
// =====================================================================
__global__ void coarse_select_kernel(const unsigned* __restrict__ hist,
                                     unsigned* __restrict__ thr) {
  const int b = threadIdx.x;
  if (b >= Bc) return;
  unsigned cum = 0, T1 = 0, above = 0;
  int found = 0;
  for (int t = 1023; t >= 0; --t) {
    const unsigned h = hist[b * 1024 + t];
    if (cum + h >= PREK) { T1 = (unsigned)t; above = cum; found = 1; break; }
    cum += h;
  }
  if (!found) { T1 = 0u; above = cum; }  // <2000 valid; bin 0 never occurs for valid keys
  thr[b * 4 + 0] = T1;
  thr[b * 4 + 1] = above;
}

// =====================================================================
// K2b: fine histogram (next 12 key bits) restricted to boundary bin.
// =====================================================================
__global__ void __launch_bounds__(256) fine_hist_kernel(
    const unsigned* __restrict__ key, const unsigned* __restrict__ thr,
    unsigned* __restrict__ fineHist) {
  __shared__ unsigned lh[4096];
  for (int i = threadIdx.x; i < 4096; i += 256) lh[i] = 0u;
  __syncthreads();
  const int blocksPerImg = Nc / 256;
  const int b = blockIdx.x / blocksPerImg;
  const int n = (blockIdx.x % blocksPerImg) * 256 + threadIdx.x;
  const unsigned k = key[(size_t)b * Nc + n];
  const unsigned T1 = thr[b * 4 + 0];
  if (k && ((k >> 20) == T1)) atomicAdd(&lh[(k >> 8) & 0xFFFu], 1u);
  __syncthreads();
  for (int i = threadIdx.x; i < 4096; i += 256)
    if (lh[i]) atomicAdd(&fineHist[b * 4096 + i], lh[i]);
}

// =====================================================================
// K2c: radix-select pass 2 — fine threshold + tie fill count.
// =====================================================================
__global__ void fine_select_kernel(const unsigned* __restrict__ fineHist,
                                   unsigned* __restrict__ thr) {
  const int b = threadIdx.x;
  if (b >= Bc) return;
  const unsigned above = thr[b * 4 + 1];
  unsigned cum = 0, T2 = 0xFFFFFFFFu, fill = 0;
  for (int t = 4095; t >= 0; --t) {
    const unsigned h = fineHist[b * 4096 + t];
    if (above + cum + h >= PREK) { T2 = (unsigned)t; fill = PREK - above - cum; break; }
    cum += h;
  }
  thr[b * 4 + 2] = T2;
  thr[b * 4 + 3] = fill;
}

// =====================================================================
// K3: compact top-2000 candidates: high region grows from 0,
// tie region fills from the top of the 2048 buffer (disjoint by construction).
// =====================================================================
__global__ void __launch_bounds__(256) compact_kernel(
    const unsigned* __restrict__ key, const unsigned* __restrict__ thr,
    unsigned* __restrict__ cnt, unsigned long long* __restrict__ cand) {
  const int blocksPerImg = Nc / 256;
  const int b = blockIdx.x / blocksPerImg;
  const int n = (blockIdx.x % blocksPerImg) * 256 + threadIdx.x;
  const unsigned k = key[(size_t)b * Nc + n];
  if (!k) return;
  const unsigned T1 = thr[b * 4 + 0], T2 = thr[b * 4 + 2], fill = thr[b * 4 + 3];
  const unsigned c1 = k >> 20, c2 = (k >> 8) & 0xFFFu;
  const bool high = (c1 > T1) || ((c1 == T1) && (c2 > T2));
  const bool tie = (c1 == T1) && (c2 == T2);
  // pack: descending sort on this => score desc, ties by ascending source index
  const unsigned long long p =
      ((unsigned long long)k << 32) | (unsigned)(~(unsigned)n);
  if (high) {
    unsigned pos = atomicAdd(&cnt[b * 2 + 0], 1u);
    if (pos < KPAD) cand[(size_t)b * KPAD + pos] = p;
  } else if (tie) {
    unsigned pos = atomicAdd(&cnt[b * 2 + 1], 1u);
    if (pos < fill) cand[(size_t)b * KPAD + (KPAD - 1 - pos)] = p;
  }
}

// =====================================================================
// K4: in-LDS bitonic sort (descending) of 2048 packed keys; gather boxes.
// =====================================================================
__global__ void __launch_bounds__(256) sort_kernel(
    const unsigned long long* __restrict__ cand, const float* __restrict__ boxes,
    const int* __restrict__ clsArr, float* __restrict__ sScore,
    float* __restrict__ sBox, int* __restrict__ sCls) {
  __shared__ unsigned long long s[KPAD];
  const int b = blockIdx.x, tid = threadIdx.x;
  for (int i = tid; i < KPAD; i += 256) s[i] = cand[(size_t)b * KPAD + i];
  __syncthreads();
  for (unsigned kk = 2; kk <= KPAD; kk <<= 1) {
    for (unsigned j = kk >> 1; j > 0; j >>= 1) {
      for (unsigned i = tid; i < KPAD; i += 256) {
        const unsigned ixj = i ^ j;
        if (ixj > i) {
          const unsigned long long x = s[i], y = s[ixj];
          const bool asc = ((i & kk) == 0);
          if (asc ? (x < y) : (x > y)) { s[i] = y; s[ixj] = x; }  // descending
        }
      }
      __syncthreads();
    }
  }
  for (int i = tid; i < KPAD; i += 256) {
    const unsigned long long p = s[i];
    const unsigned kb = (unsigned)(p >> 32);
    const size_t o = (size_t)b * KPAD + i;
    if (kb) {
      const unsigned idx = ~(unsigned)p;
      sScore[o] = __uint_as_float(kb);
      ((float4*)sBox)[o] = ((const float4*)boxes)[(size_t)b * Nc + idx];
      sCls[o] = clsArr[(size_t)b * Nc + idx];
    } else {
      sScore[o] = -1.0f;
      ((float4*)sBox)[o] = make_float4(0.f, 0.f, 0.f, 0.f);
      sCls[o] = 0;
    }
  }
}

// =====================================================================
// K5: suppression bitmask via WMMA. Each wave: 16 rows x 32 cols tile.
// Union denominator tile (area_i + area_j) computed by V_WMMA_F32_16X16X4_F32
// as a rank-2 matmul: A=[area_M,1|0,0] (16x4) x B=[1;area_N|0;0] (4x16).
// =====================================================================
__global__ void __launch_bounds__(128) iou_mask_kernel(
    const float* __restrict__ sBox, const float* __restrict__ sScore,
    unsigned* __restrict__ mask) {
  __shared__ float rowBox[4][16][4];
  __shared__ float rowSc[4][16];
  __shared__ float colBox[4][32][4];
  __shared__ float colSc[4][32];

  const int wave = threadIdx.x >> 5;
  const int lane = threadIdx.x & 31;
  const int wid = blockIdx.x * 4 + wave;
  const int b = wid / ((KPAD / 16) * (KPAD / 32));
  const int r = wid % ((KPAD / 16) * (KPAD / 32));
  const int mt = r / (KPAD / 32);
  const int nt = r % (KPAD / 32);
  const int M0 = mt * 16, N0 = nt * 32;
  const size_t sb = (size_t)b * KPAD;

  // stage tile boxes/scores into LDS
  {
    const float4 cb = ((const float4*)sBox)[sb + N0 + lane];
    colBox[wave][lane][0] = cb.x; colBox[wave][lane][1] = cb.y;
    colBox[wave][lane][2] = cb.z; colBox[wave][lane][3] = cb.w;
    colSc[wave][lane] = sScore[sb + N0 + lane];
  }
  float areaRow = 0.0f;
  if (lane < 16) {
    const float4 rb = ((const float4*)sBox)[sb + M0 + lane];
    rowBox[wave][lane][0] = rb.x; rowBox[wave][lane][1] = rb.y;
    rowBox[wave][lane][2] = rb.z; rowBox[wave][lane][3] = rb.w;
    rowSc[wave][lane] = sScore[sb + M0 + lane];
    areaRow = (rb.z - rb.x) * (rb.w - rb.y);
  }
  __syncthreads();

  // WMMA operands (f32 16x16x4): lanes 0-15 carry K=0,1; lanes 16-31 carry K=2,3 (zeros)
  v2f Amat;
  Amat.x = (lane < 16) ? areaRow : 0.0f;  // K=0: area_M   (K=2: 0)
  Amat.y = (lane < 16) ? 1.0f : 0.0f;     // K=1: 1        (K=3: 0)

  float ca0 = 0.0f, ca1 = 0.0f;
  if (lane < 16) {
    ca0 = (colBox[wave][lane][2] - colBox[wave][lane][0]) *
          (colBox[wave][lane][3] - colBox[wave][lane][1]);
    ca1 = (colBox[wave][16 + lane][2] - colBox[wave][16 + lane][0]) *
          (colBox[wave][16 + lane][3] - colBox[wave][16 + lane][1]);
  }
  v2f Bm0, Bm1;
  Bm0.x = (lane < 16) ? 1.0f : 0.0f; Bm0.y = ca0;  // K=0: 1, K=1: area_N (cols 0..15)
  Bm1.x = (lane < 16) ? 1.0f : 0.0f; Bm1.y = ca1;  // cols 16..31

  v8f D0 = {}, D1 = {};
  D0 = __builtin_amdgcn_wmma_f32_16x16x4_f32(false, Amat, false, Bm0,
                                             (short)0, D0, false, false);
  D1 = __builtin_amdgcn_wmma_f32_16x16x4_f32(false, Amat, false, Bm1,
                                             (short)0, D1, false, false);

  const int nl = lane & 15;
  const int halfOff = (lane >> 4) * 8;  // D rows: lanes 0-15 -> M=e, lanes 16-31 -> M=e+8
  const float cx1a = colBox[wave][nl][0], cy1a = colBox[wave][nl][1];
  const float cx2a = colBox[wave][nl][2], cy2a = colBox[wave][nl][3];
  const float cs0 = colSc[wave][nl];
  const float cx1b = colBox[wave][16 + nl][0], cy1b = colBox[wave][16 + nl][1];
  const float cx2b = colBox[wave][16 + nl][2], cy2b = colBox[wave][16 + nl][3];
  const float cs1 = colSc[wave][16 + nl];

#pragma unroll
  for (int e = 0; e < 8; ++e) {
    const int Mloc = e + halfOff;
    const float rx1 = rowBox[wave][Mloc][0], ry1 = rowBox[wave][Mloc][1];
    const float rx2 = rowBox[wave][Mloc][2], ry2 = rowBox[wave][Mloc][3];
    const float rs = rowSc[wave][Mloc];

    float iw = fmaxf(fminf(rx2, cx2a) - fmaxf(rx1, cx1a), 0.0f);
    float ih = fmaxf(fminf(ry2, cy2a) - fmaxf(ry1, cy1a), 0.0f);
    float inter = iw * ih;
    const bool sup0 =
        (inter > IOU_THR * fmaxf(D0[e] - inter, 1e-9f)) && (rs > cs0);

    iw = fmaxf(fminf(rx2, cx2b) - fmaxf(rx1, cx1b), 0.0f);
    ih = fmaxf(fminf(ry2, cy2b) - fmaxf(ry1, cy1b), 0.0f);
    inter = iw * ih;
    const bool sup1 =
        (inter > IOU_THR * fmaxf(D1[e] - inter, 1e-9f)) && (rs > cs1);

    const unsigned m0 = __builtin_amdgcn_ballot_w32(sup0);
    const unsigned m1 = __builtin_amdgcn_ballot_w32(sup1);
    const unsigned w_lo = (m0 & 0xFFFFu) | ((m1 & 0xFFFFu) << 16);  // row M0+e
    const unsigned w_hi = (m0 >> 16) | ((m1 >> 16) << 16);          // row M0+e+8
    if (lane == 0) {
      mask[((size_t)(b * KPAD + M0 + e)) * (KPAD / 32) + nt] = w_lo;
      mask[((size_t)(b * KPAD + M0 + e + 8)) * (KPAD / 32) + nt] = w_hi;
    }
  }
}

// =====================================================================
// K6: one wave per image — exact sequential bitmask NMS scan + outputs.
// =====================================================================
__global__ void __launch_bounds__(32) nms_scan_kernel(
    const float* __restrict__ sScore, const float* __restrict__ sBox,
    const int* __restrict__ sCls, const unsigned* __restrict__ mask,
    float* __restrict__ out) {
  __shared__ unsigned remv[KPAD / 32];  // 64 words
  __shared__ float sc[KPAD];
  __shared__ int outIdx[POSTK];
  const int b = blockIdx.x;
  const int lane = threadIdx.x;

  remv[lane] = 0u;
  remv[lane + 32] = 0u;
  for (int i = lane; i < KPAD; i += 32) sc[i] = sScore[(size_t)b * KPAD + i];
  __syncthreads();

  int cnt = 0;  // uniform across lanes
  for (int i = 0; i < PREK; ++i) {
    const unsigned r = remv[i >> 5];
    const bool kept = (sc[i] > 0.0f) && !((r >> (i & 31)) & 1u);
    if (kept) {
      if (cnt < POSTK && lane == 0) outIdx[cnt] = i;
      cnt++;
      const unsigned* mr = &mask[((size_t)(b * KPAD + i)) * (KPAD / 32)];
      remv[lane] |= mr[lane];
      remv[lane + 32] |= mr[lane + 32];
    }
    __syncthreads();
  }
  __syncthreads();

  float* const outBoxes = out;                        // [8][100][4]
  float* const outScores = out + (size_t)Bc * POSTK * 4;  // [8][100]
  float* const outCls = out + (size_t)Bc * POSTK * 5;     // [8][100]
  for (int k = lane; k < POSTK; k += 32) {
    float bx = 0.f, by = 0.f, bz = 0.f, bw = 0.f, ss = 0.f, cc = 0.f;
    if (k < cnt) {
      const int i = outIdx[k];
      const size_t o = (size_t)b * KPAD + i;
      const float4 v = ((const float4*)sBox)[o];
      bx = v.x; by = v.y; bz = v.z; bw = v.w;
      ss = sc[i];
      cc = (float)sCls[o];
    }
    ((float4*)outBoxes)[b * POSTK + k] = make_float4(bx, by, bz, bw);
    outScores[b * POSTK + k] = ss;
    outCls[b * POSTK + k] = cc;
  }
}

// =====================================================================
extern "C" void kernel_launch(void* const* d_in, const int* in_sizes, int n_in,
                              void* d_out, int out_size, void* d_ws,
                              size_t ws_size, hipStream_t stream) {
  (void)in_sizes; (void)n_in; (void)out_size; (void)ws_size;
  const float* anc = (const float*)d_in[0];  // [A*4,H,W] = 614400
  const float* ho = (const float*)d_in[1];   // [B,120,H,W]

  char* ws = (char*)d_ws;
  unsigned* hist = (unsigned*)(ws + OFF_HIST);
  unsigned* fineHist = (unsigned*)(ws + OFF_FINE);
  unsigned* cnt = (unsigned*)(ws + OFF_CNT);
  unsigned* thr = (unsigned*)(ws + OFF_THR);
  unsigned long long* cand = (unsigned long long*)(ws + OFF_CAND);
  float* boxes = (float*)(ws + OFF_BOXES);
  unsigned* key = (unsigned*)(ws + OFF_KEY);
  int* clsArr = (int*)(ws + OFF_CLS);
  float* sScore = (float*)(ws + OFF_SSC);
  float* sBox = (float*)(ws + OFF_SBOX);
  int* sCls = (int*)(ws + OFF_SCLS);
  unsigned* mask = (unsigned*)(ws + OFF_MASK);

  hipMemsetAsync(ws, 0, ZERO_BYTES, stream);  // hists/counters/candidates

  const int imgBlocks = Bc * (Nc / 256);  // 4800
  decode_kernel<<<imgBlocks, 256, 0, stream>>>(anc, ho, boxes, key, clsArr, hist);
  coarse_select_kernel<<<1, 32, 0, stream>>>(hist, thr);
  fine_hist_kernel<<<imgBlocks, 256, 0, stream>>>(key, thr, fineHist);
  fine_select_kernel<<<1, 32, 0, stream>>>(fineHist, thr);
  compact_kernel<<<imgBlocks, 256, 0, stream>>>(key, thr, cnt, cand);
  sort_kernel<<<Bc, 256, 0, stream>>>(cand, boxes, clsArr, sScore, sBox, sCls);

  const int iouWaves = Bc * (KPAD / 16) * (KPAD / 32);  // 65536 waves
  iou_mask_kernel<<<iouWaves / 4, 128, 0, stream>>>(sBox, sScore, mask);
  nms_scan_kernel<<<Bc, 32, 0, stream>>>(sScore, sBox, sCls, mask, (float*)d_out);
}